// KalmanParticleFilter_10831907520859
// MI455X (gfx1250) — compile-verified
//
#include <hip/hip_runtime.h>
#include <math.h>

// ---------------------------------------------------------------------------
// KalmanParticleFilter on MI455X (gfx1250, wave32).
// Bandwidth/latency bound fp32 pipeline; V_WMMA_F32_16X16X4_F32 offloads the
// 524288x64 sensor-distance GEMM (K=4 exactly) to the matrix pipe so it
// co-executes with the VALU top-k selection (ISA 8.4: XDL WMMA ~ TRANS).
// Sensor table is staged into LDS via the Tensor Data Mover (TENSORcnt path).
// ---------------------------------------------------------------------------

#define PN      524288
#define NBLK    2048          // PN / 256
#define NSENS   64
#define MTOP    5

typedef float v2f __attribute__((ext_vector_type(2)));
typedef float v8f __attribute__((ext_vector_type(8)));

#if defined(__has_builtin)
#if __has_builtin(__builtin_amdgcn_tensor_load_to_lds) && \
    __has_builtin(__builtin_amdgcn_s_wait_tensorcnt)
#define HAVE_TDM 1
#endif
#endif

#ifdef HAVE_TDM
typedef unsigned int u32x4 __attribute__((ext_vector_type(4)));
typedef int          i32x4 __attribute__((ext_vector_type(4)));
typedef int          i32x8 __attribute__((ext_vector_type(8)));
#endif

__constant__ static const float kMAP    = 64.0f;
static constexpr float kSPEED  = 0.5f;
static constexpr float kTPT    = 0.06283185307179586f;   // 2*pi*0.01
static constexpr float kA1     = 0.0033333333333333335f; // 0.1^2/3
static constexpr float kA2     = 0.5f * kTPT * kTPT;     // SPEED*TPT^2
static constexpr float kSXYC   = 0.0016666666666666668f + 0.25f * kTPT * kTPT; // 0.1^2/6+(0.5*TPT)^2
static constexpr float kSPHI2  = kTPT * kTPT;
static constexpr float kRDIAG  = 0.09f;                  // SENSOR_NOISE^2
static constexpr float kLWS    = 5.0f;                   // 1/(2*(0.09+0.01))
static constexpr float kNEGLOGP = -13.169796430639f;     // -log(524288)

__device__ __forceinline__ float u01hash(unsigned i) {
    unsigned h = i * 747796405u + 2891336453u;
    h = ((h >> ((h >> 28) + 4u)) ^ h) * 277803737u;
    h = (h >> 22) ^ h;
    return ((float)h + 0.5f) * 2.3283064365386963e-10f;
}

// ---------------------------------------------------------------------------
// WMMA-based nearest-sensor top-K.
// dist^2[m][n] = A[m][:] . B[:][n] with A row = (x, y, |p|^2, 1),
//               B col = (-2 sx, -2 sy, 1, |s|^2)  -> one 16x16x4 fp32 WMMA.
// A layout (32b A 16x4): lanes 0-15 give K=0,1; lanes 16-31 give K=2,3.
// B layout (4x16): VGPR0 = rows K=0 (lanes 0-15) / K=2 (lanes 16-31),
//                  VGPR1 = rows K=1 / K=3.
// D (16x16 f32): VGPR g, lane n<16 -> row g, col n; lane n+16 -> row g+8.
// D is transposed through a wave-private LDS slice so each lane recovers its
// own particle's 16 distances per sensor tile; top-K maintained in registers.
// ---------------------------------------------------------------------------
template <int K>
__device__ __forceinline__ void nearest_topk(
    float px, float py,
    const float b0[4], const float b1[4],
    float sD[][17],          // this wave's [32][17] LDS slice
    int lane,
    float bd[K], int bi[K])
{
    float nrm = px * px + py * py;
    const int m      = lane & 15;
    const bool hiH   = lane >= 16;
    const int rbase  = hiH ? 8 : 0;

    #pragma unroll
    for (int t = 0; t < 4; ++t) {                // 4 sensor tiles of 16
        #pragma unroll
        for (int Pt = 0; Pt < 2; ++Pt) {         // 2 particle tiles of 16
            const int src = 16 * Pt + m;
            float xx = __shfl(px,  src, 32);
            float yy = __shfl(py,  src, 32);
            float nn = __shfl(nrm, src, 32);
            v2f A;
            A.x = hiH ? nn   : xx;               // K=0 | K=2
            A.y = hiH ? 1.0f : yy;               // K=1 | K=3
            v2f B;
            B.x = b0[t];
            B.y = b1[t];
            v8f C = {};
            C = __builtin_amdgcn_wmma_f32_16x16x4_f32(
                    false, A, false, B, (short)0, C, false, false);
            #pragma unroll
            for (int g = 0; g < 8; ++g)
                sD[16 * Pt + rbase + g][m] = C[g];
        }
        __builtin_amdgcn_wave_barrier();         // DS in-order per wave; stop reordering
        #pragma unroll
        for (int j = 0; j < 16; ++j) {
            float d2 = sD[lane][j];
            int sid = 16 * t + j;
            if (d2 < bd[K - 1]) {                // strict < : jax tie-break (low idx)
                bd[K - 1] = d2; bi[K - 1] = sid;
                #pragma unroll
                for (int q = K - 1; q > 0; --q) {
                    if (bd[q] < bd[q - 1]) {
                        float td = bd[q]; bd[q] = bd[q - 1]; bd[q - 1] = td;
                        int   ti = bi[q]; bi[q] = bi[q - 1]; bi[q - 1] = ti;
                    }
                }
            }
        }
        __builtin_amdgcn_wave_barrier();
    }
}

// ---------------------------------------------------------------------------
// K1: fused EKF predict + update + second distance pass + log-weight.
// ---------------------------------------------------------------------------
__global__ __launch_bounds__(256) void ekf_kernel(
    const float* __restrict__ x,   const float* __restrict__ w,
    const float* __restrict__ cov, const float* __restrict__ sensors,
    const float* __restrict__ zt,  const float* __restrict__ delta,
    float* __restrict__ xnew, float* __restrict__ covnew,
    float* __restrict__ pnew, float* __restrict__ bmax)
{
    __shared__ float sSens[2 * NSENS];
    __shared__ float sDist[8][32][17];
    __shared__ float sRed[256];

    const int tid  = threadIdx.x;
    const int lane = tid & 31;
    const int wv   = tid >> 5;
    const int i    = blockIdx.x * 256 + tid;

    __builtin_prefetch(&cov[i * 9], 0, 3);       // global_prefetch_b8

#ifdef HAVE_TDM
    // Stage 128 f32 sensor values into LDS with the Tensor Data Mover:
    // one DMA issued by wave 0, tracked on TENSORcnt.
    if (wv == 0) {
        unsigned long long ga = (unsigned long long)(uintptr_t)sensors;
        unsigned ldsoff = (unsigned)(uintptr_t)(void*)sSens; // LDS aperture: offset in addr[31:0]
        u32x4 g0;
        g0.x = 1u;                                     // count=1 (valid user D#)
        g0.y = ldsoff;                                 // lds_addr (bytes)
        g0.z = (unsigned)(ga & 0xffffffffu);           // global_addr[31:0]
        g0.w = (unsigned)((ga >> 32) & 0x01ffffffu)    // global_addr[56:32]
             | (2u << 30);                             // type=2 ("image")
        i32x8 g1;
        g1[0] = (int)(2u << 16);                       // data_size=4B (bits17:16)
        g1[1] = (int)(128u << 16);                     // tensor_dim0=128 (bits79:48 lo)
        g1[2] = (int)(1u << 16);                       // tensor_dim1=1   (bits111:80 lo)
        g1[3] = (int)(128u << 16);                     // tile_dim0=128   (bits127:112)
        g1[4] = 1;                                     // tile_dim1=1     (bits143:128)
        g1[5] = 128;                                   // tensor_dim0_stride=128 (lo32)
        g1[6] = (int)(128u << 16);                     // tensor_dim1_stride=128 (lo16)
        g1[7] = 0;
        i32x4 z4 = {0, 0, 0, 0};
#if __clang_major__ >= 23
        i32x8 z8 = {0, 0, 0, 0, 0, 0, 0, 0};
        __builtin_amdgcn_tensor_load_to_lds(g0, g1, z4, z4, z8, 0);
#else
        __builtin_amdgcn_tensor_load_to_lds(g0, g1, z4, z4, 0);
#endif
        __builtin_amdgcn_s_wait_tensorcnt(0);          // s_wait_tensorcnt 0
    }
    __syncthreads();
#else
    if (tid < 2 * NSENS) sSens[tid] = sensors[tid];
    __syncthreads();
#endif

    // --- predicted state (clip x,y; theta free) ---
    float px = fminf(fmaxf(x[i * 3 + 0] + delta[i * 3 + 0], 0.0f), kMAP);
    float py = fminf(fmaxf(x[i * 3 + 1] + delta[i * 3 + 1], 0.0f), kMAP);
    float th = x[i * 3 + 2] + delta[i * 3 + 2];

    // --- per-lane B-matrix tiles for the distance GEMM ---
    float b0[4], b1[4];
    {
        const int n = lane & 15;
        #pragma unroll
        for (int t = 0; t < 4; ++t) {
            float sx = sSens[2 * (16 * t + n) + 0];
            float sy = sSens[2 * (16 * t + n) + 1];
            if (lane < 16) { b0[t] = -2.0f * sx; b1[t] = -2.0f * sy; }
            else           { b0[t] = 1.0f;      b1[t] = sx * sx + sy * sy; }
        }
    }

    // --- pass 1: top-5 nearest sensors (WMMA distance GEMM) ---
    float bd[MTOP]; int bi[MTOP];
    #pragma unroll
    for (int q = 0; q < MTOP; ++q) { bd[q] = 3.4e38f; bi[q] = 0; }
    nearest_topk<MTOP>(px, py, b0, b1, sDist[wv], lane, bd, bi);

    // --- covariance predict: covp = F cov F^T + Q (F = I + a*E02 + b*E12) ---
    float sn, cn; sincosf(th, &sn, &cn);
    const float fa = -kSPEED * sn, fb = kSPEED * cn;
    float C0[9];
    #pragma unroll
    for (int q = 0; q < 9; ++q) C0[q] = cov[i * 9 + q];
    float FC[9];
    #pragma unroll
    for (int q = 0; q < 3; ++q) {
        FC[0 + q] = C0[0 + q] + fa * C0[6 + q];
        FC[3 + q] = C0[3 + q] + fb * C0[6 + q];
        FC[6 + q] = C0[6 + q];
    }
    float Pm[3][3];
    #pragma unroll
    for (int r = 0; r < 3; ++r) {
        Pm[r][0] = FC[r * 3 + 0] + fa * FC[r * 3 + 2];
        Pm[r][1] = FC[r * 3 + 1] + fb * FC[r * 3 + 2];
        Pm[r][2] = FC[r * 3 + 2];
    }
    {
        const float s2 = sn * sn, c2 = cn * cn, s2t = 2.0f * sn * cn;
        Pm[0][0] += kA1 * c2 + kA2 * s2;
        Pm[1][1] += kA1 * s2 + kA2 * c2;
        const float sxy = kSXYC * s2t;
        Pm[0][1] += sxy; Pm[1][0] += sxy;
        Pm[0][2] += -kA2 * sn; Pm[2][0] += -kA2 * sn;
        Pm[1][2] +=  kA2 * cn; Pm[2][1] +=  kA2 * cn;
        Pm[2][2] += kSPHI2;
    }

    // --- measurement geometry ---
    float uu[MTOP], vv[MTOP], yv[MTOP];
    #pragma unroll
    for (int j = 0; j < MTOP; ++j) {
        float z  = sqrtf(bd[j]);
        float cx = sSens[2 * bi[j] + 0];
        float cy = sSens[2 * bi[j] + 1];
        float iz = 1.0f / z;
        uu[j] = (px - cx) * iz;
        vv[j] = (py - cy) * iz;
        yv[j] = zt[j] - z;
    }

    // B3 = covp H^T (3x5); S = H covp H^T + R (5x5 SPD)
    float B3[3][MTOP];
    #pragma unroll
    for (int j = 0; j < MTOP; ++j) {
        B3[0][j] = Pm[0][0] * uu[j] + Pm[0][1] * vv[j];
        B3[1][j] = Pm[1][0] * uu[j] + Pm[1][1] * vv[j];
        B3[2][j] = Pm[2][0] * uu[j] + Pm[2][1] * vv[j];
    }
    float Sm[MTOP][MTOP];
    #pragma unroll
    for (int r = 0; r < MTOP; ++r) {
        #pragma unroll
        for (int j = 0; j < MTOP; ++j)
            Sm[r][j] = uu[r] * B3[0][j] + vv[r] * B3[1][j];
        Sm[r][r] += kRDIAG;
    }

    // --- Cholesky S = L L^T (in place, lower) ---
    #pragma unroll
    for (int c = 0; c < MTOP; ++c) {
        float d = Sm[c][c];
        #pragma unroll
        for (int k2 = 0; k2 < MTOP; ++k2)
            if (k2 < c) d -= Sm[c][k2] * Sm[c][k2];
        float ld = sqrtf(fmaxf(d, 1e-20f));
        Sm[c][c] = ld;
        float inv = 1.0f / ld;
        #pragma unroll
        for (int r = 0; r < MTOP; ++r) {
            if (r > c) {
                float vvv = Sm[r][c];
                #pragma unroll
                for (int k2 = 0; k2 < MTOP; ++k2)
                    if (k2 < c) vvv -= Sm[r][k2] * Sm[c][k2];
                Sm[r][c] = vvv * inv;
            }
        }
    }

    // --- K = covp H^T S^-1 via two triangular solves per row ---
    float Kk[3][MTOP];
    #pragma unroll
    for (int r = 0; r < 3; ++r) {
        float tsol[MTOP];
        #pragma unroll
        for (int a = 0; a < MTOP; ++a) {
            float vvv = B3[r][a];
            #pragma unroll
            for (int k2 = 0; k2 < MTOP; ++k2)
                if (k2 < a) vvv -= Sm[a][k2] * tsol[k2];
            tsol[a] = vvv / Sm[a][a];
        }
        #pragma unroll
        for (int a = MTOP - 1; a >= 0; --a) {
            float vvv = tsol[a];
            #pragma unroll
            for (int k2 = 0; k2 < MTOP; ++k2)
                if (k2 > a) vvv -= Sm[k2][a] * tsol[k2];
            tsol[a] = vvv / Sm[a][a];
            Kk[r][a] = tsol[a];
        }
    }

    // --- state + covariance update ---
    float xr[3] = { px, py, th };
    float xn[3];
    #pragma unroll
    for (int r = 0; r < 3; ++r) {
        float acc = xr[r];
        #pragma unroll
        for (int j = 0; j < MTOP; ++j) acc += Kk[r][j] * yv[j];
        xn[r] = acc;
    }
    float HP[MTOP][3];
    #pragma unroll
    for (int j = 0; j < MTOP; ++j) {
        #pragma unroll
        for (int c = 0; c < 3; ++c)
            HP[j][c] = uu[j] * Pm[0][c] + vv[j] * Pm[1][c];
    }
    float cvn[3][3];
    #pragma unroll
    for (int r = 0; r < 3; ++r)
        #pragma unroll
        for (int c = 0; c < 3; ++c) {
            float acc = Pm[r][c];
            #pragma unroll
            for (int j = 0; j < MTOP; ++j) acc -= Kk[r][j] * HP[j][c];
            cvn[r][c] = acc;
        }

    // --- pass 2: top-2 distances from x_new (WMMA again) -> log weight ---
    float bd2[2]; int bi2[2];
    bd2[0] = bd2[1] = 3.4e38f; bi2[0] = bi2[1] = 0;
    nearest_topk<2>(xn[0], xn[1], b0, b1, sDist[wv], lane, bd2, bi2);
    float d0 = zt[0] - sqrtf(bd2[0]);
    float d1 = zt[1] - sqrtf(bd2[1]);
    float p  = w[i] - (d0 * d0 + d1 * d1) * kLWS;

    // --- store staged results ---
    #pragma unroll
    for (int r = 0; r < 3; ++r) xnew[i * 3 + r] = xn[r];
    #pragma unroll
    for (int r = 0; r < 3; ++r)
        #pragma unroll
        for (int c = 0; c < 3; ++c) covnew[i * 9 + r * 3 + c] = cvn[r][c];
    pnew[i] = p;

    // --- block max of p for log-softmax ---
    sRed[tid] = p; __syncthreads();
    for (int s3 = 128; s3 > 0; s3 >>= 1) {
        if (tid < s3) sRed[tid] = fmaxf(sRed[tid], sRed[tid + s3]);
        __syncthreads();
    }
    if (tid == 0) bmax[blockIdx.x] = sRed[0];
}

// K2: global max of 2048 block maxima -> scal[0]
__global__ __launch_bounds__(256) void max_kernel(const float* __restrict__ bmax,
                                                  float* __restrict__ scal)
{
    __shared__ float r[256];
    int tid = threadIdx.x;
    float m = -3.4e38f;
    #pragma unroll
    for (int j = 0; j < 8; ++j) m = fmaxf(m, bmax[tid * 8 + j]);
    r[tid] = m; __syncthreads();
    for (int s = 128; s > 0; s >>= 1) {
        if (tid < s) r[tid] = fmaxf(r[tid], r[tid + s]);
        __syncthreads();
    }
    if (tid == 0) scal[0] = r[0];
}

// K3: per-block sum of exp(p-gmax) and exp(p-gmax)^2
__global__ __launch_bounds__(256) void sums_kernel(const float* __restrict__ pnew,
                                                   const float* __restrict__ scal,
                                                   float* __restrict__ bsum,
                                                   float* __restrict__ bsum2)
{
    __shared__ float r1[256], r2[256];
    int tid = threadIdx.x;
    int i = blockIdx.x * 256 + tid;
    float g = scal[0];
    float e = expf(pnew[i] - g);
    r1[tid] = e; r2[tid] = e * e; __syncthreads();
    for (int s = 128; s > 0; s >>= 1) {
        if (tid < s) { r1[tid] += r1[tid + s]; r2[tid] += r2[tid + s]; }
        __syncthreads();
    }
    if (tid == 0) { bsum[blockIdx.x] = r1[0]; bsum2[blockIdx.x] = r2[0]; }
}

// K4: exclusive scan of 2048 block sums, totals, logZ, n_eff, resample flag
__global__ __launch_bounds__(256) void scan_kernel(const float* __restrict__ bsum,
                                                   const float* __restrict__ bsum2,
                                                   float* __restrict__ scanE,
                                                   float* __restrict__ scal)
{
    __shared__ float ws[256];
    __shared__ float r2[256];
    int tid = threadIdx.x;
    float loc[8];
    float s = 0.0f, s2 = 0.0f;
    #pragma unroll
    for (int j = 0; j < 8; ++j) {
        loc[j] = bsum[tid * 8 + j];
        s += loc[j];
        s2 += bsum2[tid * 8 + j];
    }
    ws[tid] = s; r2[tid] = s2; __syncthreads();
    for (int off = 1; off < 256; off <<= 1) {        // Hillis-Steele inclusive
        float v = ws[tid];
        float add = (tid >= off) ? ws[tid - off] : 0.0f;
        __syncthreads();
        ws[tid] = v + add;
        __syncthreads();
    }
    float run = ws[tid] - s;                          // thread-exclusive prefix
    #pragma unroll
    for (int j = 0; j < 8; ++j) { scanE[tid * 8 + j] = run; run += loc[j]; }
    for (int st = 128; st > 0; st >>= 1) {
        if (tid < st) r2[tid] += r2[tid + st];
        __syncthreads();
    }
    if (tid == 0) {
        float total = ws[255];
        float gmax  = scal[0];
        float logZ  = gmax + logf(total);
        float neff  = (total * total) / r2[0];
        scal[1] = total;
        scal[2] = logZ;
        scal[3] = (neff <= 0.25f * (float)PN) ? 1.0f : 0.0f;
        scal[4] = neff;
    }
}

// K5: emit outputs; if resampling, inverse-CDF sample via block-CDF binary
// search (staged in LDS) + in-block walk, then gather.
__global__ __launch_bounds__(256) void out_kernel(const float* __restrict__ xnew,
                                                  const float* __restrict__ covnew,
                                                  const float* __restrict__ pnew,
                                                  const float* __restrict__ scanE,
                                                  const float* __restrict__ scal,
                                                  float* __restrict__ out)
{
    __shared__ float sScan[NBLK];
    int tid = threadIdx.x;
    int i = blockIdx.x * 256 + tid;
    for (int j = tid; j < NBLK; j += 256) sScan[j] = scanE[j];
    __syncthreads();

    float doF = scal[3];
    int idx = i;
    float wout;
    if (doF > 0.5f) {
        float total = scal[1], gmax = scal[0];
        float uq = u01hash((unsigned)i) * total;
        int lo = 0, hi = NBLK;                         // sScan[0] == 0 <= uq
        while (hi - lo > 1) {
            int mid = (lo + hi) >> 1;
            if (sScan[mid] <= uq) lo = mid; else hi = mid;
        }
        float acc = sScan[lo];
        int jj;
        for (jj = 0; jj < 256; ++jj) {
            acc += expf(pnew[lo * 256 + jj] - gmax);
            if (uq < acc) break;
        }
        if (jj == 256) jj = 255;
        idx = lo * 256 + jj;
        wout = kNEGLOGP;
    } else {
        wout = pnew[i] - scal[2];
    }
    #pragma unroll
    for (int r = 0; r < 3; ++r) out[i * 3 + r] = xnew[idx * 3 + r];
    out[3 * PN + i] = wout;
    #pragma unroll
    for (int q = 0; q < 9; ++q) out[4 * PN + i * 9 + q] = covnew[idx * 9 + q];
}

extern "C" void kernel_launch(void* const* d_in, const int* in_sizes, int n_in,
                              void* d_out, int out_size, void* d_ws, size_t ws_size,
                              hipStream_t stream) {
    const float* x       = (const float*)d_in[0];
    const float* w       = (const float*)d_in[1];
    const float* cov     = (const float*)d_in[2];
    const float* sensors = (const float*)d_in[3];
    const float* zt      = (const float*)d_in[4];
    const float* delta   = (const float*)d_in[5];
    float* ws   = (float*)d_ws;
    float* xnew   = ws;
    float* covnew = ws + (size_t)3 * PN;
    float* pnew   = ws + (size_t)12 * PN;
    float* bmax   = ws + (size_t)13 * PN;
    float* bsum   = bmax + NBLK;
    float* bsum2  = bsum + NBLK;
    float* scanE  = bsum2 + NBLK;
    float* scal   = scanE + NBLK;
    float* out = (float*)d_out;

    ekf_kernel <<<NBLK, 256, 0, stream>>>(x, w, cov, sensors, zt, delta,
                                          xnew, covnew, pnew, bmax);
    max_kernel <<<1,    256, 0, stream>>>(bmax, scal);
    sums_kernel<<<NBLK, 256, 0, stream>>>(pnew, scal, bsum, bsum2);
    scan_kernel<<<1,    256, 0, stream>>>(bsum, bsum2, scanE, scal);
    out_kernel <<<NBLK, 256, 0, stream>>>(xnew, covnew, pnew, scanE, scal, out);
}